// LennardJonesEnergy_68332929679956
// MI455X (gfx1250) — compile-verified
//
#include <hip/hip_runtime.h>
#include <stdint.h>

// ---------------------------------------------------------------------------
// LJ edge-energy scatter: memory-bound streaming kernel for MI455X (gfx1250).
// Index streams (51.2 MB, the only real HBM traffic) are DMA'd into LDS with
// the Tensor Data Mover (tensor_load_to_lds, TENSORcnt), double-buffered per
// block. Position gathers (1.2 MB, L2-resident) use b96 vector loads; the
// scatter-add uses hardware global_atomic_add_f32 into the 0.4 MB accumulator.
// HBM floor: ~53 MB / 23.3 TB/s ~= 2.3 us.
// ---------------------------------------------------------------------------

typedef uint32_t v4u __attribute__((ext_vector_type(4)));
typedef int      v8i __attribute__((ext_vector_type(8)));
typedef int      v4i __attribute__((ext_vector_type(4)));

#define LJ_TILE   1024       // edges per tile (4KB per index array)
#define LJ_BLOCK  256        // 8 wave32s

#if __has_builtin(__builtin_amdgcn_tensor_load_to_lds) && __has_builtin(__builtin_amdgcn_s_wait_tensorcnt)
#define LJ_HAS_TDM 1
#else
#define LJ_HAS_TDM 0
#endif

struct F3 { float x, y, z; };   // 12B, 4B-aligned -> global_load_b96

__global__ void lj_zero_kernel(float* __restrict__ out, int n) {
    int i = blockIdx.x * blockDim.x + threadIdx.x;
    if (i < n) out[i] = 0.0f;
}

#if LJ_HAS_TDM
// Issue a 1-D TDM copy of `count` 4-byte elements: global -> LDS.
// D# layout per CDNA5 ISA ch.7 (CDNA5 ISA); 6-arg builtin form.
__device__ __forceinline__ void tdm_load_1d(uint32_t lds_byte_addr,
                                            const void* gptr,
                                            uint32_t count) {
    uint64_t ga = (uint64_t)(uintptr_t)gptr;
    v4u g0;
    g0.x = 1u;                                    // count=1 (valid), user mode
    g0.y = lds_byte_addr;                         // bits 63:32  lds_addr
    g0.z = (uint32_t)ga;                          // bits 95:64  global_addr lo
    g0.w = (uint32_t)(ga >> 32) | 0x80000000u;    // global_addr hi | type=2
    v8i g1;
    g1[0] = 0x00020000;                  // workgroup_mask=0, data_size=2 (4B)
    g1[1] = (int)(count << 16);          // tensor_dim0[15:0]  @ bits 63:48
    g1[2] = (int)((count >> 16) | (1u << 16)); // tensor_dim0 hi, tensor_dim1=1
    g1[3] = (int)(count << 16);          // tile_dim0 @ bits 127:112
    g1[4] = 0;                           // tile_dim1=0 (1-D), tile_dim2=0
    g1[5] = (int)count;                  // tensor_dim0_stride lo
    g1[6] = 0;
    g1[7] = 0;
    v4i z4 = {0, 0, 0, 0};               // groups 2/3 unused (<=2D tensor)
    v8i z8 = {0, 0, 0, 0, 0, 0, 0, 0};
    __builtin_amdgcn_tensor_load_to_lds(g0, g1, z4, z4, z8, 0);
}
#endif

__device__ __forceinline__ void lj_edge_body(const F3* __restrict__ p3,
                                             float* __restrict__ out,
                                             int s, int r) {
    F3 pr = p3[r];                         // global_load_b96 (L2 hit)
    F3 ps = p3[s];                         // global_load_b96 (L2 hit)
    float dx = pr.x - ps.x;
    float dy = pr.y - ps.y;
    float dz = pr.z - ps.z;
    float r2   = dx * dx + dy * dy + dz * dz;
    float inv2 = 1.0f / r2;                // (sigma/r)^2, sigma = 1
    float inv6 = inv2 * inv2 * inv2;       // (sigma/r)^6
    float e    = 2.0f * (inv6 * inv6 - inv6);
    unsafeAtomicAdd(&out[r], e);           // -> global_atomic_add_f32 (L2)
}

__global__ void __launch_bounds__(LJ_BLOCK)
lj_edge_kernel(const float* __restrict__ pos,
               const int*   __restrict__ senders,
               const int*   __restrict__ receivers,
               float*       __restrict__ out,
               int n_edges) {
    __shared__ int smem[2][2][LJ_TILE];   // [buffer][senders/receivers][edge]

    const F3* __restrict__ p3 = (const F3*)pos;
    const int ntiles = (n_edges + LJ_TILE - 1) / LJ_TILE;
    const int t0 = blockIdx.x;

#if LJ_HAS_TDM
    const uint32_t lds_base = (uint32_t)(uintptr_t)(&smem[0][0][0]);

    // Prologue: kick tile t0 into buffer 0 (wave 0 only; TDM ignores EXEC,
    // one wave issues each DMA exactly once).
    if (threadIdx.x == 0) {
        long base = (long)t0 * LJ_TILE;
        long rem  = (long)n_edges - base;
        uint32_t cnt = rem < LJ_TILE ? (uint32_t)rem : (uint32_t)LJ_TILE;
        tdm_load_1d(lds_base + 0u,           senders   + base, cnt);
        tdm_load_1d(lds_base + 4u * LJ_TILE, receivers + base, cnt);
    }
#endif

    int phase = 0;
    for (int t = t0; t < ntiles; t += gridDim.x, phase ^= 1) {
        long base = (long)t * LJ_TILE;
        long rem  = (long)n_edges - base;
        int  cnt  = rem < LJ_TILE ? (int)rem : LJ_TILE;

#if LJ_HAS_TDM
        // Kick next tile's DMAs into the other buffer, then wait for the
        // current tile's 2 DMAs (TDM ops complete in order per wave).
        if (threadIdx.x == 0) {
            int tn = t + gridDim.x;
            if (tn < ntiles) {
                long nbase = (long)tn * LJ_TILE;
                long nrem  = (long)n_edges - nbase;
                uint32_t ncnt = nrem < LJ_TILE ? (uint32_t)nrem : (uint32_t)LJ_TILE;
                uint32_t boff = (uint32_t)((phase ^ 1) * 2 * LJ_TILE * 4);
                tdm_load_1d(lds_base + boff,                senders   + nbase, ncnt);
                tdm_load_1d(lds_base + boff + 4u * LJ_TILE, receivers + nbase, ncnt);
                __builtin_amdgcn_s_wait_tensorcnt(2);
            } else {
                __builtin_amdgcn_s_wait_tensorcnt(0);
            }
        }
        __syncthreads();
#else
        // Fallback staging: cooperative coalesced copy into LDS.
        for (int i = threadIdx.x; i < cnt; i += LJ_BLOCK) {
            smem[phase][0][i] = senders[base + i];
            smem[phase][1][i] = receivers[base + i];
        }
        __syncthreads();
#endif

        if (cnt == LJ_TILE) {
            // Full tile: unrolled so 8 b96 gathers are in flight per wave
            // before any dependent VALU work.
            int ss[LJ_TILE / LJ_BLOCK], rr[LJ_TILE / LJ_BLOCK];
#pragma unroll
            for (int k = 0; k < LJ_TILE / LJ_BLOCK; ++k) {
                int i = threadIdx.x + k * LJ_BLOCK;
                ss[k] = smem[phase][0][i];
                rr[k] = smem[phase][1][i];
            }
#pragma unroll
            for (int k = 0; k < LJ_TILE / LJ_BLOCK; ++k) {
                lj_edge_body(p3, out, ss[k], rr[k]);
            }
        } else {
            for (int i = threadIdx.x; i < cnt; i += LJ_BLOCK) {
                lj_edge_body(p3, out, smem[phase][0][i], smem[phase][1][i]);
            }
        }
        __syncthreads();  // protect smem[phase] before it is re-filled
    }
}

extern "C" void kernel_launch(void* const* d_in, const int* in_sizes, int n_in,
                              void* d_out, int out_size, void* d_ws, size_t ws_size,
                              hipStream_t stream) {
    const float* pos       = (const float*)d_in[0];
    const int*   senders   = (const int*)d_in[1];
    const int*   receivers = (const int*)d_in[2];
    float*       out       = (float*)d_out;

    const int n_nodes = in_sizes[0] / 3;
    const int n_edges = in_sizes[1];

    // d_out is poisoned by the harness: zero the accumulator first.
    lj_zero_kernel<<<(n_nodes + 255) / 256, 256, 0, stream>>>(out, n_nodes);

    if (n_edges > 0) {
        int ntiles = (n_edges + LJ_TILE - 1) / LJ_TILE;
        int blocks = ntiles < 1024 ? ntiles : 1024;   // ~6 tiles/block -> pipeline
        lj_edge_kernel<<<blocks, LJ_BLOCK, 0, stream>>>(pos, senders, receivers,
                                                        out, n_edges);
    }
}